// ScatterAttention_87686052315506
// MI455X (gfx1250) — compile-verified
//
#include <hip/hip_runtime.h>
#include <math.h>

#define D 128

typedef float v2f  __attribute__((ext_vector_type(2)));
typedef float v8f  __attribute__((ext_vector_type(8)));
typedef float f32x4 __attribute__((ext_vector_type(4)));

// ---------- float <-> ordered-uint key for atomic max on signed floats ----------
__device__ __forceinline__ unsigned floatKey(float f) {
    unsigned b = __float_as_uint(f);
    return (b & 0x80000000u) ? ~b : (b | 0x80000000u);   // monotone map; all real keys > 0
}
__device__ __forceinline__ float keyToFloat(unsigned k) {
    return __uint_as_float((k & 0x80000000u) ? (k ^ 0x80000000u) : ~k);
}

// ---------------- K0: tiny precompute  M = Wq*Wk^T, v = Wk*bq, u = Wq*bk, s0 = bq.bk --------
__global__ void sa_precompute(const float* __restrict__ Wq, const float* __restrict__ bq,
                              const float* __restrict__ Wk, const float* __restrict__ bk,
                              float* __restrict__ M, float* __restrict__ v,
                              float* __restrict__ u, float* __restrict__ s0) {
    const int tid = threadIdx.x;                       // 256 threads, one block
    for (int i = tid; i < D * D; i += 256) {
        const int dp = i >> 7, d = i & 127;
        float acc = 0.f;
        #pragma unroll 8
        for (int p = 0; p < D; ++p) acc = fmaf(Wq[dp * D + p], Wk[d * D + p], acc);
        M[i] = acc;
    }
    if (tid < D) {
        float av = 0.f, au = 0.f;
        #pragma unroll 8
        for (int p = 0; p < D; ++p) {
            av = fmaf(bq[p], Wk[tid * D + p], av);
            au = fmaf(Wq[tid * D + p], bk[p], au);
        }
        v[tid] = av; u[tid] = au;
    }
    if (tid == 0) {
        float a = 0.f;
        for (int p = 0; p < D; ++p) a = fmaf(bq[p], bk[p], a);
        *s0 = a;
    }
}

// ---------------- K0b: zero seg_sum / seg_max keys / attn output -------------------
__global__ void sa_init(float* __restrict__ segsum, unsigned* __restrict__ segmax,
                        float* __restrict__ attn, int Q) {
    const int i = blockIdx.x * 256 + threadIdx.x;
    if (i < Q) segsum[i] = 0.f;
    if (i < Q) segmax[i] = 0u;                      // key 0 == "-infinity" sentinel
    const int total = Q * D;
    for (int j = i; j < total; j += gridDim.x * 256) attn[j] = 0.f;
}

// ---------------- K1: t = queries @ M + v ; c = queries . u + s0  (fp32 WMMA) ------
// 8 waves / block; wave w computes 16-row tile rt = blockIdx.x*8 + w over all 128 cols.
__global__ __launch_bounds__(256) void sa_tproj(const float* __restrict__ queries,
                                                const float* __restrict__ Mmat,
                                                const float* __restrict__ vvec,
                                                const float* __restrict__ uvec,
                                                const float* __restrict__ s0p,
                                                float* __restrict__ t, float* __restrict__ c,
                                                int nTiles) {
    __shared__ float Ms[D * D];                     // 64 KB: whole M resident per WGP
    const int tid = threadIdx.x;
    {
        const f32x4* src = (const f32x4*)Mmat;
        f32x4* dst = (f32x4*)Ms;
        for (int i = tid; i < (D * D) / 4; i += 256) dst[i] = src[i];
    }
    __syncthreads();

    const int wave  = tid >> 5;
    const int lane  = tid & 31;
    const int halfK = lane >> 4;                    // lanes 16-31 hold K+2 per ISA layout
    const int m     = lane & 15;
    const int rt    = blockIdx.x * 8 + wave;
    if (rt >= nTiles) return;

    // ---- preload A fragments: row (rt*16+m), K pairs at k = 4s + 2*halfK + {0,1} ----
    v2f a[32];
    const float* arow = queries + (size_t)(rt * 16 + m) * D + 2 * halfK;
    #pragma unroll
    for (int s = 0; s < 32; ++s) { a[s].x = arow[4 * s]; a[s].y = arow[4 * s + 1]; }

    // ---- c[q] = queries[q] . u + s0, reduced across the two half-waves ----
    {
        float part = 0.f;
        const float* up = uvec + 2 * halfK;
        #pragma unroll
        for (int s = 0; s < 32; ++s)
            part = fmaf(a[s].x, up[4 * s], fmaf(a[s].y, up[4 * s + 1], part));
        part += __shfl_xor(part, 16, 32);
        if (halfK == 0) c[rt * 16 + m] = part + *s0p;
    }

    // ---- 8 N-tiles x 32 K-steps of v_wmma_f32_16x16x4_f32 ----
    for (int nt = 0; nt < 8; ++nt) {
        const int col = nt * 16 + m;                 // B lane n = lane%16
        const float* bp = Ms + (size_t)(2 * halfK) * D + col;
        v8f acc = {};
        #pragma unroll
        for (int s = 0; s < 32; ++s) {
            v2f b; b.x = bp[(4 * s) * D]; b.y = bp[(4 * s + 1) * D];
            acc = __builtin_amdgcn_wmma_f32_16x16x4_f32(
                false, a[s], false, b, (short)0, acc, false, false);
        }
        const float vb = vvec[col];
        float* trow = t + (size_t)(rt * 16 + 8 * halfK) * D + col;  // C row = r + 8*halfK
        #pragma unroll
        for (int r = 0; r < 8; ++r) trow[(size_t)r * D] = acc[r] + vb;
    }
}

// ---------------- K2: probs[e] = scale*(sv[e].t[idx] + c[idx]); atomic seg-max -----
__global__ __launch_bounds__(256) void sa_probs(const float* __restrict__ sv,
                                                const int* __restrict__ idx,
                                                const float* __restrict__ t,
                                                const float* __restrict__ c,
                                                float* __restrict__ probs,
                                                unsigned* __restrict__ segmax, int E) {
    const int wave = threadIdx.x >> 5, lane = threadIdx.x & 31;
    const int e = blockIdx.x * 8 + wave;
    if (e >= E) return;
    const int q = idx[e];
    const f32x4 aa = ((const f32x4*)(sv + (size_t)e * D))[lane];
    const f32x4 bb = ((const f32x4*)(t  + (size_t)q * D))[lane];   // gather: t is L2-resident
    float p = aa.x * bb.x + aa.y * bb.y + aa.z * bb.z + aa.w * bb.w;
    p += __shfl_xor(p, 16, 32);
    p += __shfl_xor(p,  8, 32);
    p += __shfl_xor(p,  4, 32);
    p += __shfl_xor(p,  2, 32);
    p += __shfl_xor(p,  1, 32);
    if (lane == 0) {
        const float scale = 0.088388347648318447f;   // 128^-0.5
        const float pr = (p + c[q]) * scale;
        probs[e] = pr;
        atomicMax(segmax + q, floatKey(pr));
    }
}

// ---------------- K3: seg_sum += exp(probs - seg_max) ------------------------------
__global__ void sa_expsum(const float* __restrict__ probs, const int* __restrict__ idx,
                          const unsigned* __restrict__ segmax, float* __restrict__ segsum, int E) {
    const int e = blockIdx.x * 256 + threadIdx.x;
    if (e >= E) return;
    const int q = idx[e];
    const float ex = __expf(probs[e] - keyToFloat(segmax[q]));
    __hip_atomic_fetch_add(segsum + q, ex, __ATOMIC_RELAXED, __HIP_MEMORY_SCOPE_AGENT);
}

// ---------------- K4: scores + weighted scatter-sum of sv into attn ----------------
__global__ __launch_bounds__(256) void sa_scatter(const float* __restrict__ sv,
                                                  const int* __restrict__ idx,
                                                  const float* __restrict__ probs,
                                                  const unsigned* __restrict__ segmax,
                                                  const float* __restrict__ segsum,
                                                  float* __restrict__ scores,
                                                  float* __restrict__ attn, int E) {
    const int wave = threadIdx.x >> 5, lane = threadIdx.x & 31;
    const int e = blockIdx.x * 8 + wave;
    if (e >= E) return;
    const int q = idx[e];
    const float s = __expf(probs[e] - keyToFloat(segmax[q])) / segsum[q];
    if (lane == 0) scores[e] = s;
    const f32x4 aa = ((const f32x4*)(sv + (size_t)e * D))[lane];
    float* dst = attn + (size_t)q * D + lane * 4;    // attn region is L2-resident
    __hip_atomic_fetch_add(dst + 0, aa.x * s, __ATOMIC_RELAXED, __HIP_MEMORY_SCOPE_AGENT);
    __hip_atomic_fetch_add(dst + 1, aa.y * s, __ATOMIC_RELAXED, __HIP_MEMORY_SCOPE_AGENT);
    __hip_atomic_fetch_add(dst + 2, aa.z * s, __ATOMIC_RELAXED, __HIP_MEMORY_SCOPE_AGENT);
    __hip_atomic_fetch_add(dst + 3, aa.w * s, __ATOMIC_RELAXED, __HIP_MEMORY_SCOPE_AGENT);
}

// ----------------------------------- launch ---------------------------------------
extern "C" void kernel_launch(void* const* d_in, const int* in_sizes, int n_in,
                              void* d_out, int out_size, void* d_ws, size_t ws_size,
                              hipStream_t stream) {
    const float* sv  = (const float*)d_in[0];
    const int*   idx = (const int*)  d_in[1];
    const float* qrs = (const float*)d_in[2];
    const float* Wq  = (const float*)d_in[3];
    const float* bq  = (const float*)d_in[4];
    const float* Wk  = (const float*)d_in[5];
    const float* bk  = (const float*)d_in[6];

    const int E = in_sizes[1];          // 800000
    const int Q = in_sizes[2] / D;      // 50000

    // workspace layout (floats)
    float* ws      = (float*)d_ws;
    float* wM      = ws;                                  // 128*128
    float* wv      = wM + D * D;                          // 128
    float* wu      = wv + D;                              // 128
    float* ws0     = wu + D;                              // 1 (+pad)
    float* wt      = ws0 + 256;                           // Q*128
    float* wc      = wt + (size_t)Q * D;                  // Q
    float* wprobs  = wc + Q;                              // E
    unsigned* wmax = (unsigned*)(wprobs + E);             // Q
    float* wsum    = (float*)(wmax + Q);                  // Q

    float* out_scores = (float*)d_out;                    // [E]
    float* out_attn   = out_scores + E;                   // [Q*128]

    sa_precompute<<<1, 256, 0, stream>>>(Wq, bq, Wk, bk, wM, wv, wu, ws0);

    {
        const int n = Q * (D + 2);
        sa_init<<<(n + 255) / 256, 256, 0, stream>>>(wsum, wmax, out_attn, Q);
    }

    const int nTiles = Q / 16;                            // 3125
    sa_tproj<<<(nTiles + 7) / 8, 256, 0, stream>>>(qrs, wM, wv, wu, ws0, wt, wc, nTiles);

    sa_probs<<<(E + 7) / 8, 256, 0, stream>>>(sv, idx, wt, wc, wprobs, wmax, E);

    sa_expsum<<<(E + 255) / 256, 256, 0, stream>>>(wprobs, idx, wmax, wsum, E);

    sa_scatter<<<(E + 7) / 8, 256, 0, stream>>>(sv, idx, wprobs, wmax, wsum,
                                                out_scores, out_attn, E);
}